// GINEN_18580028523178
// MI455X (gfx1250) — compile-verified
//
#include <hip/hip_runtime.h>

typedef __attribute__((ext_vector_type(2))) float v2f;
typedef __attribute__((ext_vector_type(8))) float v8f;

#define DFEAT 128

// ---------------------------------------------------------------------------
// Degree: deg[t] += 1 for every edge target
// ---------------------------------------------------------------------------
__global__ __launch_bounds__(256) void gine_deg_kernel(
    const int* __restrict__ tgt, float* __restrict__ deg, int E) {
  int e = blockIdx.x * blockDim.x + threadIdx.x;
  if (e < E) atomicAdd(&deg[tgt[e]], 1.0f);
}

// ---------------------------------------------------------------------------
// Edge phase: msg = relu(h[src] + ea[e]*lin_w + lin_b); agg[tgt] += msg
// One wave per edge, each lane handles 4 consecutive channels.
// h (25.6 MB) and agg (25.6 MB) are resident in the 192 MB L2, so the
// random gather + atomic scatter is on-chip traffic, not HBM.
// ---------------------------------------------------------------------------
__global__ __launch_bounds__(256) void gine_scatter_kernel(
    const float* __restrict__ h, const int* __restrict__ src,
    const int* __restrict__ tgt, const float* __restrict__ ea,
    const float* __restrict__ lw, const float* __restrict__ lb,
    float* __restrict__ agg, int E) {
  int t = blockIdx.x * blockDim.x + threadIdx.x;
  int lane = t & 31;
  int e = t >> 5;
  int c = lane * 4;
  if (e >= E) return;
  const float4 lwv = *(const float4*)(lw + c);
  const float4 lbv = *(const float4*)(lb + c);
  int s = src[e];
  int d = tgt[e];
  float w = ea[e];
  const float4 hv = *(const float4*)(h + (size_t)s * DFEAT + c);
  float4 m;
  m.x = fmaxf(hv.x + w * lwv.x + lbv.x, 0.0f);
  m.y = fmaxf(hv.y + w * lwv.y + lbv.y, 0.0f);
  m.z = fmaxf(hv.z + w * lwv.z + lbv.z, 0.0f);
  m.w = fmaxf(hv.w + w * lwv.w + lbv.w, 0.0f);
  float* ap = agg + (size_t)d * DFEAT + c;
  atomicAdd(ap + 0, m.x);
  atomicAdd(ap + 1, m.y);
  atomicAdd(ap + 2, m.z);
  atomicAdd(ap + 3, m.w);
}

// ---------------------------------------------------------------------------
// out = agg/deg + (1+eps[l])*h    (in place on agg)
// ---------------------------------------------------------------------------
__global__ __launch_bounds__(256) void gine_norm_eps_kernel(
    float* __restrict__ agg, const float* __restrict__ h,
    const float* __restrict__ deg, const float* __restrict__ eps, int l,
    int total) {
  int i = blockIdx.x * blockDim.x + threadIdx.x;
  if (i >= total) return;
  int node = i >> 7;  // i / 128
  float scale = 1.0f + eps[l];
  agg[i] = agg[i] / deg[node] + scale * h[i];
}

// ---------------------------------------------------------------------------
// W [K,128] row-major  ->  Wt [128,K]  (so B fragments become contiguous b64)
// ---------------------------------------------------------------------------
__global__ __launch_bounds__(256) void gine_transpose_w_kernel(
    const float* __restrict__ W, float* __restrict__ Wt, int K) {
  int i = blockIdx.x * blockDim.x + threadIdx.x;
  if (i < K * DFEAT) {
    int k = i >> 7;
    int c = i & 127;
    Wt[(size_t)c * K + k] = W[i];
  }
}

// ---------------------------------------------------------------------------
// WMMA GEMM: C[nrows,128] = (opt relu)( concat_s(A_s) @ W + bias )
//   A_s : NSEG row-major [nrows,128] segments (virtual concat along K)
//   Wt  : [128, NSEG*128] = transpose of W  (lane B-fragment = 1x b64 load)
// Block = 256 threads = 8 waves; block owns 16 rows x 128 cols.
// The 16x128 A tiles are staged into LDS with GLOBAL_LOAD_ASYNC_TO_LDS_B128
// (CDNA5 async copy: memory -> LDS without touching VGPRs, retired on
// ASYNCcnt), stride padded to 132 floats so per-lane ds_load_b64 hits
// distinct banks. All 8 waves then share the tile instead of re-fetching
// it 8x from VMEM. K walked in steps of 4 via V_WMMA_F32_16X16X4_F32
// (full fp32 precision; total GEMM work ~33 GFLOP, far below the matrix
// roof, so no bf16 downcast needed).
// ---------------------------------------------------------------------------
template <int NSEG>
__global__ __launch_bounds__(256) void gine_wmma_gemm_kernel(
    const float* __restrict__ A0, const float* __restrict__ A1,
    const float* __restrict__ A2, const float* __restrict__ A3,
    const float* __restrict__ Wt, const float* __restrict__ bias,
    float* __restrict__ C, int nrows, int doRelu) {
  constexpr int LDST = 132;              // padded LDS row stride (floats)
  __shared__ float Atile[NSEG * 16 * LDST];

  const int tid = threadIdx.x;
  const int wave = tid >> 5;             // 0..7 -> 16-column tile
  const int lane = tid & 31;
  const int hi = lane >> 4;              // lane half selects K pair
  const int lo = lane & 15;
  const int row0 = blockIdx.x * 16;
  const int col = wave * 16 + lo;
  const int rowsIn = min(16, nrows - row0);

  // ---- async stage of the NSEG 16x128 A tiles into LDS ----
#pragma unroll
  for (int s = 0; s < NSEG; ++s) {
    const float* A = (s == 0) ? A0 : (s == 1) ? A1 : (s == 2) ? A2 : A3;
#pragma unroll
    for (int t = 0; t < 2; ++t) {
      int idx = tid + t * 256;           // float4 index within 16x128 tile
      int r = idx >> 5;                  // tile row 0..15
      int c4 = idx & 31;                 // float4 column
      float* dst = &Atile[(s * 16 + r) * LDST + c4 * 4];
      if (r < rowsIn) {
        unsigned lds_off = (unsigned)(unsigned long long)dst;
        unsigned long long gaddr =
            (unsigned long long)(A + (size_t)(row0 + r) * DFEAT + c4 * 4);
        // async global -> LDS copy (16 B/lane), tracked by ASYNCcnt
        asm volatile("global_load_async_to_lds_b128 %0, %1, off"
                     :
                     : "v"(lds_off), "v"(gaddr)
                     : "memory");
      } else {
        *(float4*)dst = make_float4(0.0f, 0.0f, 0.0f, 0.0f);
      }
    }
  }
  asm volatile("s_wait_asynccnt 0x0" ::: "memory");
  __syncthreads();

  // ---- main WMMA loop ----
  v8f acc = {};
#pragma unroll
  for (int s = 0; s < NSEG; ++s) {
    const float* wtp = Wt + (size_t)col * (NSEG * DFEAT) + s * DFEAT;
    const float* atp = &Atile[(s * 16 + lo) * LDST];
#pragma unroll
    for (int kt = 0; kt < DFEAT / 4; ++kt) {
      int k = kt * 4 + 2 * hi;
      v2f a = *(const v2f*)(atp + k);    // ds_load_b64 (A fragment)
      v2f b = *(const v2f*)(wtp + k);    // global_load_b64 (B fragment)
      acc = __builtin_amdgcn_wmma_f32_16x16x4_f32(
          /*neg_a=*/false, a, /*neg_b=*/false, b,
          /*c_mod=*/(short)0, acc, /*reuse_a=*/false, /*reuse_b=*/false);
    }
  }

  // ---- epilogue: bias (+relu) + store ----
  float bcol = bias[col];
#pragma unroll
  for (int v = 0; v < 8; ++v) {
    int r = row0 + v + 8 * hi;           // C layout: VGPR v holds rows v, v+8
    if (r < nrows) {
      float val = acc[v] + bcol;
      if (doRelu) val = fmaxf(val, 0.0f);
      C[(size_t)r * DFEAT + col] = val;
    }
  }
}

// ---------------------------------------------------------------------------
// Host-side orchestration (graph-capture safe: only async ops on `stream`)
// ---------------------------------------------------------------------------
extern "C" void kernel_launch(void* const* d_in, const int* in_sizes, int n_in,
                              void* d_out, int out_size, void* d_ws,
                              size_t ws_size, hipStream_t stream) {
  const float* x     = (const float*)d_in[0];   // [N,128]
  const int*   ei    = (const int*)d_in[1];     // [2,E]
  const float* ea    = (const float*)d_in[2];   // [E,1]
  const float* lin_w = (const float*)d_in[3];   // [3,1,128]
  const float* lin_b = (const float*)d_in[4];   // [3,128]
  const float* eps   = (const float*)d_in[5];   // [3]
  const float* w1    = (const float*)d_in[6];   // [3,128,128]
  const float* b1    = (const float*)d_in[7];   // [3,128]
  const float* w2    = (const float*)d_in[8];   // [3,128,128]
  const float* b2    = (const float*)d_in[9];   // [3,128]
  const float* fw    = (const float*)d_in[10];  // [512,128]
  const float* fb    = (const float*)d_in[11];  // [128]
  float* out = (float*)d_out;

  const int n = in_sizes[0] / DFEAT;
  const int E = in_sizes[2];
  const int* src = ei;
  const int* tgt = ei + E;

  // workspace carve-up (floats)
  const size_t nd = (size_t)n * DFEAT;
  float* ws  = (float*)d_ws;
  float* deg = ws;               // [n]  (slot padded to 65536)
  float* agg = ws + 65536;       // [n,128], also normalized in place
  float* mid = agg + nd;         // [n,128]
  float* h1  = mid + nd;         // [n,128]
  float* h2  = h1 + nd;          // [n,128]
  float* h3  = h2 + nd;          // [n,128]
  float* wt  = h3 + nd;          // [<=512*128] transposed weight staging
  float* hbufs[3] = {h1, h2, h3};

  const int rowBlocks = (n + 15) / 16;
  const int edgeThreadBlocks = (int)(((size_t)E * 32 + 255) / 256);
  const int tposeBlocks128 = (DFEAT * DFEAT + 255) / 256;

  hipMemsetAsync(deg, 0, (size_t)n * sizeof(float), stream);
  gine_deg_kernel<<<(E + 255) / 256, 256, 0, stream>>>(tgt, deg, E);

  const float* h = x;
  for (int l = 0; l < 3; ++l) {
    hipMemsetAsync(agg, 0, nd * sizeof(float), stream);
    gine_scatter_kernel<<<edgeThreadBlocks, 256, 0, stream>>>(
        h, src, tgt, ea, lin_w + l * DFEAT, lin_b + l * DFEAT, agg, E);
    gine_norm_eps_kernel<<<(int)((nd + 255) / 256), 256, 0, stream>>>(
        agg, h, deg, eps, l, (int)nd);

    // mid = relu(agg @ w1[l] + b1[l])
    gine_transpose_w_kernel<<<tposeBlocks128, 256, 0, stream>>>(
        w1 + (size_t)l * DFEAT * DFEAT, wt, DFEAT);
    gine_wmma_gemm_kernel<1><<<rowBlocks, 256, 0, stream>>>(
        agg, agg, agg, agg, wt, b1 + l * DFEAT, mid, n, 1);

    // h_{l+1} = mid @ w2[l] + b2[l]
    gine_transpose_w_kernel<<<tposeBlocks128, 256, 0, stream>>>(
        w2 + (size_t)l * DFEAT * DFEAT, wt, DFEAT);
    gine_wmma_gemm_kernel<1><<<rowBlocks, 256, 0, stream>>>(
        mid, mid, mid, mid, wt, b2 + l * DFEAT, hbufs[l], n, 0);

    h = hbufs[l];
  }

  // out = [x, h1, h2, h3] @ final_w + final_b   (virtual concat, K = 512)
  gine_transpose_w_kernel<<<(4 * DFEAT * DFEAT + 255) / 256, 256, 0, stream>>>(
      fw, wt, 4 * DFEAT);
  gine_wmma_gemm_kernel<4><<<rowBlocks, 256, 0, stream>>>(
      x, h1, h2, h3, wt, fb, out, n, 0);
}